// ChamferLoss_37031208026602
// MI455X (gfx1250) — compile-verified
//
#include <hip/hip_runtime.h>
#include <stdint.h>

#define BATCH 16
#define MPTS  4096
#define NPTS  4096
#define CH    6
#define CEPS  1e-8f

typedef __attribute__((ext_vector_type(2))) float v2f;
typedef __attribute__((ext_vector_type(8))) float v8f;
typedef unsigned long long u64;

#define INIT_KEY 0x7F80000000000000ull  // {+inf bits, idx 0}

__device__ __forceinline__ u64 umin64(u64 a, u64 b) { return a < b ? a : b; }

// ---------------- kernel 0: init ws ----------------
__global__ void chamfer_init(u64* __restrict__ gbwd, float* __restrict__ acc) {
    int i = blockIdx.x * blockDim.x + threadIdx.x;
    if (i < 2) acc[i] = 0.0f;
    if (i < BATCH * NPTS) gbwd[i] = INIT_KEY;
}

// ---------------- kernel 1: fused WMMA distance + dual argmin ----------------
// grid = BATCH * (MPTS/128), block = 256 (8 waves, each wave owns a 16-row m-tile)
__global__ __launch_bounds__(256)
void chamfer_main(const float* __restrict__ pred6, const float* __restrict__ gt6,
                  const float* __restrict__ thresh_p,
                  u64* __restrict__ gbwd, float* __restrict__ acc) {
    __shared__ float s_gt[NPTS * 4];   // 64 KB: {x, y, z, ||g||^2} per gt point
    __shared__ u64   s_bwd[NPTS];      // 32 KB: per-workgroup backward packed mins

    const int tid    = threadIdx.x;
    const int wgPerB = MPTS / 128;
    const int b      = blockIdx.x / wgPerB;
    const int mblk   = blockIdx.x % wgPerB;
    const int wave   = tid >> 5;
    const int lane   = tid & 31;
    const int half   = lane >> 4;      // 0: K0/K1 rows (x,y), 1: K2/K3 rows (z,0)
    const int l16    = lane & 15;
    const int m_base = mblk * 128 + wave * 16;

    const float* pred = pred6 + (size_t)b * CH * MPTS;
    const float* gt   = gt6   + (size_t)b * CH * NPTS;

    // ---- cooperative staging: gt points + squared norms into LDS, init bwd mins ----
    for (int i = tid; i < NPTS; i += 256) {
        float x = gt[i], y = gt[NPTS + i], z = gt[2 * NPTS + i];
        float4 q; q.x = x; q.y = y; q.z = z; q.w = x * x + y * y + z * z;
        *(float4*)&s_gt[i * 4] = q;    // ds_store_b128
        s_bwd[i] = INIT_KEY;
    }
    __syncthreads();

    // ---- A fragment: 16x4 f32 (x,y,z,0), pre-scaled by -2 so WMMA yields -2*dot ----
    const int am = m_base + l16;
    float ax0 = pred[am];                    // K0 row value (x) for this lane's m
    float ax2 = pred[2 * MPTS + am];         // K2 row value (z)
    float ax1 = pred[1 * MPTS + am];         // K1 row value (y)
    float a0  = half ? ax2 : ax0;
    float a1  = half ? 0.0f : ax1;
    float tp  = a0 * a0 + a1 * a1;
    float p2l = tp + __shfl_xor(tp, 16, 32); // ||pred_{m_base+l16}||^2 (once, outside loop)

    v2f a; a.x = -2.0f * a0; a.y = -2.0f * a1;

    float p2reg[8];                          // p2 indexed like C rows: M = v + 8*half
    unsigned mlo[8];                         // packed-key low word for backward (m index)
#pragma unroll
    for (int v = 0; v < 8; ++v) {
        p2reg[v] = __shfl(p2l, v + (half ? 8 : 0), 32);
        mlo[v]   = (unsigned)(m_base + v + 8 * half);
    }

    u64 fpk[8];                              // forward running packed min over n
#pragma unroll
    for (int v = 0; v < 8; ++v) fpk[v] = INIT_KEY;

#pragma unroll 2
    for (int nb = 0; nb < NPTS; nb += 16) {
        const int gn = nb + l16;
        float4 q = *(const float4*)&s_gt[gn * 4];   // one ds_load_b128, no divergence
        float b0 = half ? q.z : q.x;
        float b1 = half ? 0.0f : q.y;
        float g2 = q.w;

        v2f bb; bb.x = b0; bb.y = b1;
        v8f c = {0.f, 0.f, 0.f, 0.f, 0.f, 0.f, 0.f, 0.f};
        // c[v] = -2 * dot(pred_{m_base+v+8*half}, gt_{gn})
        c = __builtin_amdgcn_wmma_f32_16x16x4_f32(false, a, false, bb,
                                                  (short)0, c, false, false);
        u64 bpk[8];
#pragma unroll
        for (int v = 0; v < 8; ++v) {
            float d2  = (p2reg[v] + g2) + c[v];
            u64 key   = (u64)__float_as_uint(fmaxf(d2, 0.0f)) << 32;
            fpk[v]    = umin64(fpk[v], key | (unsigned)gn);     // v_min_u64
            bpk[v]    = key | mlo[v];
        }
        // backward: depth-3 v_min_u64 tree over the tile's 8 m rows in this lane
#pragma unroll
        for (int st = 4; st >= 1; st >>= 1)
#pragma unroll
            for (int v = 0; v < st; ++v) bpk[v] = umin64(bpk[v], bpk[v + st]);
        // all 32 lanes hit s_bwd[gn]; lanes L and L+16 combine in the LDS atomic unit
        atomicMin(&s_bwd[gn], bpk[0]);                          // ds_min_u64
    }

    // ---- forward finalize: butterfly min across the 16-lane half, then gather+loss ----
#pragma unroll
    for (int off = 1; off < 16; off <<= 1) {
#pragma unroll
        for (int v = 0; v < 8; ++v)
            fpk[v] = umin64(fpk[v], __shfl_xor(fpk[v], off, 32));
    }
    const float thresh = thresh_p[0];
    float fsum = 0.0f;
    if (l16 < 8) {                     // 16 active lanes, one m each
        int m    = m_base + l16 + half * 8;
        int nidx = (int)(fpk[l16] & 0xffffffffu);
        float px = pred[m], py = pred[MPTS + m], pz = pred[2 * MPTS + m];
        float gx = s_gt[nidx * 4], gy = s_gt[nidx * 4 + 1], gz = s_gt[nidx * 4 + 2];
        float dx = gx - px, dy = gy - py, dz = gz - pz;
        float e  = sqrtf(dx * dx + dy * dy + dz * dz + CEPS);
        fsum = fmaxf(e - thresh, 0.0f);
    }
#pragma unroll
    for (int off = 16; off >= 1; off >>= 1) fsum += __shfl_xor(fsum, off, 32);
    if (lane == 0) atomicAdd(&acc[0], fsum);

    // ---- flush backward staging to global packed atomic-min buffer ----
    __syncthreads();
    for (int i = tid; i < NPTS; i += 256)
        atomicMin(&gbwd[(size_t)b * NPTS + i], s_bwd[i]);       // global_atomic_min_u64
}

// ---------------- kernel 2: backward gather + loss ----------------
__global__ __launch_bounds__(256)
void chamfer_bwd(const float* __restrict__ pred6, const float* __restrict__ gt6,
                 const float* __restrict__ thresh_p,
                 const u64* __restrict__ gbwd, float* __restrict__ acc) {
    const int idx = blockIdx.x * 256 + threadIdx.x;   // grid sized exactly B*N/256
    const int b = idx / NPTS, n = idx % NPTS;
    const float* pred = pred6 + (size_t)b * CH * MPTS;
    const float* gt   = gt6   + (size_t)b * CH * NPTS;
    const int m = (int)(gbwd[idx] & 0xffffffffu);
    float px = pred[m], py = pred[MPTS + m], pz = pred[2 * MPTS + m];
    float gx = gt[n],   gy = gt[NPTS + n],  gz = gt[2 * NPTS + n];
    float dx = px - gx, dy = py - gy, dz = pz - gz;
    float e = sqrtf(dx * dx + dy * dy + dz * dz + CEPS);
    float r = fmaxf(e - thresh_p[0], 0.0f);
#pragma unroll
    for (int off = 16; off >= 1; off >>= 1) r += __shfl_xor(r, off, 32);
    if ((threadIdx.x & 31) == 0) atomicAdd(&acc[1], r);
}

// ---------------- kernel 3: combine means ----------------
__global__ void chamfer_final(const float* __restrict__ acc, float* __restrict__ out) {
    out[0] = acc[0] * (1.0f / (float)(BATCH * MPTS)) +
             acc[1] * (1.0f / (float)(BATCH * NPTS));
}

extern "C" void kernel_launch(void* const* d_in, const int* in_sizes, int n_in,
                              void* d_out, int out_size, void* d_ws, size_t ws_size,
                              hipStream_t stream) {
    (void)in_sizes; (void)n_in; (void)out_size; (void)ws_size;
    const float* pred6    = (const float*)d_in[0];
    const float* gt6      = (const float*)d_in[1];
    const float* thresh_p = (const float*)d_in[2];
    float* out = (float*)d_out;

    u64*   gbwd = (u64*)d_ws;                                        // B*N u64 = 512 KB
    float* acc  = (float*)((char*)d_ws + (size_t)BATCH * NPTS * 8);  // 2 floats

    chamfer_init<<<(BATCH * NPTS + 255) / 256, 256, 0, stream>>>(gbwd, acc);
    chamfer_main<<<BATCH * (MPTS / 128), 256, 0, stream>>>(pred6, gt6, thresh_p, gbwd, acc);
    chamfer_bwd<<<(BATCH * NPTS) / 256, 256, 0, stream>>>(pred6, gt6, thresh_p, gbwd, acc);
    chamfer_final<<<1, 1, 0, stream>>>(acc, out);
}